// FwdRealTimeNonLinearRTUs_54863912239692
// MI455X (gfx1250) — compile-verified
//
#include <hip/hip_runtime.h>

#define B_  128
#define D_  256
#define H_  256

typedef __attribute__((ext_vector_type(2))) float v2f;
typedef __attribute__((ext_vector_type(4))) float v4f;
typedef __attribute__((ext_vector_type(8))) float v8f;

// ---------------------------------------------------------------------------
// Kernel 0: per-column (H) parameter precompute -> d_ws
// Layout in P: [0]=g, [1]=phi, [2]=norm, [3]=d_g_w_r, [4]=d_phi_w_r,
//              [5]=d_g_w_th, [6]=d_phi_w_th, [7]=d_norm_w_r   (each H floats)
// ---------------------------------------------------------------------------
__global__ __launch_bounds__(H_)
void rtu_params_kernel(const float* __restrict__ r_param,
                       const float* __restrict__ theta_param,
                       float* __restrict__ P) {
    const int j = threadIdx.x;
    const float er  = __expf(r_param[j]);
    const float r   = __expf(-er);
    const float th  = __expf(theta_param[j]);
    const float g   = r * __cosf(th);
    const float phi = r * __sinf(th);
    const float nrm = __builtin_sqrtf(1.0f - r * r);
    P[0 * H_ + j] = g;
    P[1 * H_ + j] = phi;
    P[2 * H_ + j] = nrm;
    P[3 * H_ + j] = -er * g;          // d_g_w_r
    P[4 * H_ + j] = -er * phi;        // d_phi_w_r
    P[5 * H_ + j] = -th * phi;        // d_g_w_th
    P[6 * H_ + j] =  th * g;          // d_phi_w_th
    P[7 * H_ + j] =  er * r * r / nrm;// d_norm_w_r
}

// ---------------------------------------------------------------------------
// Kernel 1: WMMA f32 GEMM (w1x = X@W1, w2x = X@W2) fused with the small
// elementwise part: c1/c2, relu, ngm0..ngm3.  One wave per 16x16 tile.
// A-matrix layout (16x4 f32): lanes 0-15 hold K={k0,k0+1}, lanes 16-31 hold
// K={k0+2,k0+3} (VGPR0=even K, VGPR1=odd K).  B/C/D row-striped across lanes.
// ---------------------------------------------------------------------------
__global__ __launch_bounds__(256)
void rtu_gemm_fused_kernel(const float* __restrict__ X,
                           const float* __restrict__ W1,
                           const float* __restrict__ W2,
                           const float* __restrict__ hc1,
                           const float* __restrict__ hc2,
                           const float* __restrict__ gm0,
                           const float* __restrict__ gm1,
                           const float* __restrict__ gm2,
                           const float* __restrict__ gm3,
                           const float* __restrict__ P,
                           float* __restrict__ out) {
    const int lane = threadIdx.x & 31;
    const int wave = (blockIdx.x * blockDim.x + threadIdx.x) >> 5; // 0..127
    const int tm   = wave >> 4;          // 0..7  (B tiles)
    const int tn   = wave & 15;          // 0..15 (H tiles)
    const int m0   = tm * 16;
    const int n0   = tn * 16;
    const int half = lane >> 4;          // 0 or 1
    const int l15  = lane & 15;

    v8f acc1 = {};
    v8f acc2 = {};

    const int arow = m0 + l15;           // A row for this lane
    const int bcol = n0 + l15;           // B col for this lane

    for (int k0 = 0; k0 < D_; k0 += 4) {
        const int ka = k0 + 2 * half;    // this lane's K pair
        v2f a = *(const v2f*)(X + arow * D_ + ka);
        v2f b1, b2;
        b1.x = W1[(ka + 0) * H_ + bcol];
        b1.y = W1[(ka + 1) * H_ + bcol];
        b2.x = W2[(ka + 0) * H_ + bcol];
        b2.y = W2[(ka + 1) * H_ + bcol];
        acc1 = __builtin_amdgcn_wmma_f32_16x16x4_f32(
            false, a, false, b1, (short)0, acc1, false, false);
        acc2 = __builtin_amdgcn_wmma_f32_16x16x4_f32(
            false, a, false, b2, (short)0, acc2, false, false);
    }

    // Per-column parameters (column n = n0 + l15, duplicated across halves)
    const int n = n0 + l15;
    const float g   = P[0 * H_ + n];
    const float phi = P[1 * H_ + n];
    const float nrm = P[2 * H_ + n];
    const float dgr = P[3 * H_ + n];
    const float dpr = P[4 * H_ + n];
    const float dgt = P[5 * H_ + n];
    const float dpt = P[6 * H_ + n];
    const float dnr = P[7 * H_ + n];

    const size_t BH = (size_t)B_ * H_;
    float* h1o = out;
    float* h2o = out + BH;
    float* o0  = out + 2 * BH;
    float* o1  = out + 3 * BH;
    float* o2  = out + 4 * BH;
    float* o3  = out + 5 * BH;

#pragma unroll
    for (int v = 0; v < 8; ++v) {
        const int m   = m0 + v + 8 * half;   // C/D layout: VGPR v, lane half
        const int idx = m * H_ + n;
        const float w1 = acc1[v];
        const float w2 = acc2[v];
        const float h1c = hc1[idx];
        const float h2c = hc2[idx];
        const float c1 = g * h1c - phi * h2c + nrm * w1;
        const float c2 = g * h2c + phi * h1c + nrm * w2;
        const float d1 = c1 > 0.0f ? 1.0f : 0.0f;
        const float d2 = c2 > 0.0f ? 1.0f : 0.0f;
        h1o[idx] = d1 * c1;                  // relu(c1)
        h2o[idx] = d2 * c2;                  // relu(c2)
        const float m0v = gm0[idx];
        const float m1v = gm1[idx];
        const float m2v = gm2[idx];
        const float m3v = gm3[idx];
        o0[idx] = d1 * (dgr * h1c + g * m0v - dpr * h2c - phi * m1v + dnr * w1);
        o1[idx] = d2 * (dgr * h2c + g * m1v + dpr * h1c + phi * m0v + dnr * w2);
        o2[idx] = d1 * (dgt * h1c + g * m2v - dpt * h2c - phi * m3v);
        o3[idx] = d2 * (dgt * h2c + g * m3v + dpt * h1c + phi * m2v);
    }
}

// ---------------------------------------------------------------------------
// Kernel 2: the bandwidth kernel. ngm4..ngm7 over (B, D_IN, H).
// One thread per 4 contiguous j; b128 loads + NON-TEMPORAL b128 stores.
// Inputs stay RT so the 134 MB gm read-set remains resident in the 192 MB L2
// across graph replays; the 134 MB output stream bypasses L2 allocation (NT),
// so steady-state HBM traffic is dominated by the write stream alone.
// d1/d2 are recovered from h1_post/h2_post > 0  (== c>0 since h = relu(c)).
// ---------------------------------------------------------------------------
__global__ __launch_bounds__(256)
void rtu_trace_kernel(const float* __restrict__ X,
                      const float* __restrict__ gm4,
                      const float* __restrict__ gm5,
                      const float* __restrict__ gm6,
                      const float* __restrict__ gm7,
                      const float* __restrict__ P,
                      const float* __restrict__ out_head, // h1_post, h2_post
                      float* __restrict__ o4,
                      float* __restrict__ o5,
                      float* __restrict__ o6,
                      float* __restrict__ o7) {
    const int tid = blockIdx.x * blockDim.x + threadIdx.x;
    const int jv  = (tid & 63) << 2;     // j in 0..252 step 4
    const int bi  = tid >> 6;            // b*D_IN + i
    const int b   = bi >> 8;             // D_IN == 256

    const size_t base = (size_t)bi * H_ + jv;
    const v4f m4 = *(const v4f*)(gm4 + base);
    const v4f m5 = *(const v4f*)(gm5 + base);
    const v4f m6 = *(const v4f*)(gm6 + base);
    const v4f m7 = *(const v4f*)(gm7 + base);

    const int hb = b * H_ + jv;
    const v4f h1 = *(const v4f*)(out_head + hb);
    const v4f h2 = *(const v4f*)(out_head + (size_t)B_ * H_ + hb);
    const v4f gV = *(const v4f*)(P + 0 * H_ + jv);
    const v4f pV = *(const v4f*)(P + 1 * H_ + jv);
    const v4f nV = *(const v4f*)(P + 2 * H_ + jv);
    const float xv = X[bi];              // x_t[b, i], broadcast over j

    v4f r4, r5, r6, r7;
#define RTU_COMP(c)                                                 \
    {                                                               \
        const float d1 = h1.c > 0.0f ? 1.0f : 0.0f;                 \
        const float d2 = h2.c > 0.0f ? 1.0f : 0.0f;                 \
        const float a1 = d1 * gV.c, p1 = d1 * pV.c, q1 = d1 * nV.c; \
        const float a2 = d2 * gV.c, p2 = d2 * pV.c, q2 = d2 * nV.c; \
        r4.c = a1 * m4.c - p1 * m5.c + q1 * xv;                     \
        r5.c = a2 * m5.c + p2 * m4.c;                               \
        r6.c = a1 * m6.c - p1 * m7.c;                               \
        r7.c = a2 * m7.c + p2 * m6.c + q2 * xv;                     \
    }
    RTU_COMP(x) RTU_COMP(y) RTU_COMP(z) RTU_COMP(w)
#undef RTU_COMP

    __builtin_nontemporal_store(r4, (v4f*)(o4 + base));
    __builtin_nontemporal_store(r5, (v4f*)(o5 + base));
    __builtin_nontemporal_store(r6, (v4f*)(o6 + base));
    __builtin_nontemporal_store(r7, (v4f*)(o7 + base));
}

// ---------------------------------------------------------------------------
extern "C" void kernel_launch(void* const* d_in, const int* in_sizes, int n_in,
                              void* d_out, int out_size, void* d_ws, size_t ws_size,
                              hipStream_t stream) {
    const float* x   = (const float*)d_in[0];
    const float* hc1 = (const float*)d_in[1];
    const float* hc2 = (const float*)d_in[2];
    const float* gm0 = (const float*)d_in[3];
    const float* gm1 = (const float*)d_in[4];
    const float* gm2 = (const float*)d_in[5];
    const float* gm3 = (const float*)d_in[6];
    const float* gm4 = (const float*)d_in[7];
    const float* gm5 = (const float*)d_in[8];
    const float* gm6 = (const float*)d_in[9];
    const float* gm7 = (const float*)d_in[10];
    const float* rp  = (const float*)d_in[11];
    const float* tp  = (const float*)d_in[12];
    const float* w1  = (const float*)d_in[13];
    const float* w2  = (const float*)d_in[14];

    float* out = (float*)d_out;
    float* P   = (float*)d_ws;           // 8 * H_ floats = 8 KB

    const size_t BH  = (size_t)B_ * H_;
    const size_t BDH = (size_t)B_ * D_ * H_;

    rtu_params_kernel<<<1, H_, 0, stream>>>(rp, tp, P);

    // 128 waves = 16 blocks x 8 waves (wave32)
    rtu_gemm_fused_kernel<<<16, 256, 0, stream>>>(
        x, w1, w2, hc1, hc2, gm0, gm1, gm2, gm3, P, out);

    // B * D_IN * (H/4) threads = 2,097,152 -> 8192 blocks of 256
    rtu_trace_kernel<<<(B_ * D_ * (H_ / 4)) / 256, 256, 0, stream>>>(
        x, gm4, gm5, gm6, gm7, P, out,
        out + 6 * BH, out + 6 * BH + BDH,
        out + 6 * BH + 2 * BDH, out + 6 * BH + 3 * BDH);
}